// Net_29102698398058
// MI455X (gfx1250) — compile-verified
//
#include <hip/hip_runtime.h>
#include <hip/hip_bf16.h>

// CDNA5 / gfx1250, wave32.
typedef __attribute__((ext_vector_type(2))) float v2f;
typedef __attribute__((ext_vector_type(8))) float v8f;

#define DIM 128   // embedding dim
#define NN  64    // nodes per graph
#define EE  16    // edge types (= candidate count)
#define VDD 300   // raw vec dim

// ---------------------------------------------------------------------------
// Kernel 1: leaf[i] = data_vecs[types[i], data[i]] @ data_weights[types[i]] + b
// 64 blocks x 128 threads; the 300-float vec is staged in LDS once per node.
// ---------------------------------------------------------------------------
__global__ void leaf_kernel(const int* __restrict__ data,
                            const int* __restrict__ types,
                            const float* __restrict__ data_vecs,
                            const float* __restrict__ data_weights,
                            const float* __restrict__ data_biases,
                            float* __restrict__ leaf, long V) {
  __shared__ float svec[VDD];
  const int i = blockIdx.x;
  const int n = threadIdx.x;
  const int t = types[i];
  const long d = (long)data[i];
  const float* vec = data_vecs + ((long)t * V + d) * VDD;
  for (int k = n; k < VDD; k += DIM) svec[k] = vec[k];
  __syncthreads();
  const float* W = data_weights + (long)t * VDD * DIM;
  float acc = data_biases[t * DIM + n];
  for (int k = 0; k < VDD; ++k) acc = fmaf(svec[k], W[k * DIM + n], acc);
  leaf[i * DIM + n] = acc;
}

// ---------------------------------------------------------------------------
// Kernel 2: one workgroup per graph (256 threads = 8 wave32).
//  Setup : parents/depth/hc/path/candidate list; prefetch path W matrices.
//  acc   : initialized to leaf (so e = acc everywhere; node `pos` is a leaf
//          with no children, so acc[pos] stays == leaf[pos]).
//  msg63 : msg63[e] = leaf[pos] @ W_e + b_e computed in-block (2 edges/wave).
//  Phase A: candidate-independent bottom-up DP, skipping message propagation
//           for nodes on the path pos -> root. float4 W loads, ds_add_f32.
//  Phase B: all 16 candidates batched up the path: chain of [16x128]@[128x128]
//           GEMMs on V_WMMA_F32_16X16X4_F32 (wave w owns N-tile w).
//  Phase C: scoring.
// ---------------------------------------------------------------------------
__global__ void tree_kernel(const int* __restrict__ graphs,
                            const int* __restrict__ edges,
                            const int* __restrict__ pos_p,
                            const float* __restrict__ leaf,
                            const float* __restrict__ edge_weights,
                            const float* __restrict__ edge_biases,
                            const float* __restrict__ sew,
                            const float* __restrict__ seb,
                            const float* __restrict__ sdw,
                            const float* __restrict__ sdb,
                            float* __restrict__ out) {
  __shared__ float acc[NN][DIM];          // 32 KB (holds leaf + child messages)
  __shared__ float ecand[2][EE][DIM];     // 16 KB double buffer
  __shared__ float msg63s[EE][DIM];       // 8 KB
  __shared__ int par[NN], depth[NN], hc[NN], onpath[NN], path[NN];
  __shared__ int cedge[EE];
  __shared__ int meta[3];                 // path_len, maxdepth, pos

  const int j    = blockIdx.x;
  const int tid  = threadIdx.x;
  const int wave = tid >> 5;
  const int lane = tid & 31;

  if (tid == 0) {
    const int pos = pos_p[0];
    for (int i = 0; i < NN; ++i) par[i] = i + graphs[j * NN + i];
    depth[0] = 0;
    int md = 0;
    for (int i = 1; i < NN; ++i) {
      depth[i] = depth[par[i]] + 1;
      if (depth[i] > md) md = depth[i];
    }
    for (int i = 0; i < NN; ++i) { hc[i] = 0; onpath[i] = 0; }
    for (int i = 1; i < NN; ++i) hc[par[i]] = 1;
    int p = pos, idx = 0;
    for (;;) { path[idx++] = p; onpath[p] = 1; if (p == 0) break; p = par[p]; }
    meta[0] = idx; meta[1] = md; meta[2] = pos;
    // candidate edge ids: slot0 = correct (graph 0) or leftover (others),
    // slots 1..15 = all edges != correct, in order; leftover = last of those.
    const int correct = edges[pos];
    int cnt = 1;
    for (int e = 0; e < EE; ++e) if (e != correct) cedge[cnt++] = e;
    const int leftover = cedge[EE - 1];
    cedge[0] = (j == 0) ? correct : leftover;
  }
  __syncthreads();

  const int path_len = meta[0];
  const int maxdepth = meta[1];
  const int pos      = meta[2];

  // Warm WGP$/L2 with the path's edge-weight matrices (the serial-chain
  // critical path of Phase B). 64KB each; 256B (2 lines) per thread.
  for (int s = 1; s + 1 < path_len; ++s) {
    const char* Wp = (const char*)(edge_weights + edges[path[s]] * DIM * DIM);
    const int off = tid * 256;
    __builtin_prefetch(Wp + off, 0, 1);        // global_prefetch_b8
    __builtin_prefetch(Wp + off + 128, 0, 1);
  }

  // acc := leaf  (coalesced float4)
  for (int idx = tid; idx < NN * DIM / 4; idx += 256)
    ((float4*)acc)[idx] = ((const float4*)leaf)[idx];
  __syncthreads();

  // msg63s[e] = leaf[pos] @ W_e + b_e  (graph-independent; 2 edges per wave)
  for (int e = wave * 2; e < wave * 2 + 2; ++e) {
    const float4* W4 = (const float4*)(edge_weights + e * DIM * DIM);
    float4 m = ((const float4*)(edge_biases + e * DIM))[lane];
    for (int k = 0; k < DIM; ++k) {
      const float ek = acc[pos][k];
      const float4 w = W4[k * (DIM / 4) + lane];
      m.x = fmaf(ek, w.x, m.x);
      m.y = fmaf(ek, w.y, m.y);
      m.z = fmaf(ek, w.z, m.z);
      m.w = fmaf(ek, w.w, m.w);
    }
    ((float4*)msg63s[e])[lane] = m;
  }
  __syncthreads();

  // -------- Phase A: shared bottom-up DP (skip on-path nodes) --------
  for (int d = maxdepth; d >= 1; --d) {
    int cnt = 0;
    for (int i = 1; i < NN; ++i) {
      if (depth[i] == d && !onpath[i]) {
        if ((cnt & 7) == wave) {
          const int e = edges[i];
          const float4* W4 = (const float4*)(edge_weights + e * DIM * DIM);
          const int pr = par[i];
          const int relu = hc[i];
          float4 m = ((const float4*)(edge_biases + e * DIM))[lane];
          for (int k = 0; k < DIM; ++k) {
            float ek = acc[i][k];
            if (relu) ek = fmaxf(ek, 0.f);
            const float4 w = W4[k * (DIM / 4) + lane];   // global_load_b128
            m.x = fmaf(ek, w.x, m.x);
            m.y = fmaf(ek, w.y, m.y);
            m.z = fmaf(ek, w.z, m.z);
            m.w = fmaf(ek, w.w, m.w);
          }
          atomicAdd(&acc[pr][4 * lane + 0], m.x);        // ds_add_f32
          atomicAdd(&acc[pr][4 * lane + 1], m.y);
          atomicAdd(&acc[pr][4 * lane + 2], m.z);
          atomicAdd(&acc[pr][4 * lane + 3], m.w);
        }
        cnt++;
      }
    }
    __syncthreads();
  }

  // -------- Phase B: candidate-batched chain up the path (WMMA) --------
  const int p1 = path[1];
  for (int idx = tid; idx < EE * DIM; idx += 256) {
    const int c = idx >> 7, k = idx & (DIM - 1);
    const float v = acc[p1][k] + msg63s[cedge[c]][k];
    ecand[0][c][k] = fmaxf(v, 0.f);   // p1 always has children -> relu
  }
  __syncthreads();

  int buf = 0;
  const int half = lane >> 4;        // 0: lanes 0-15, 1: lanes 16-31
  const int l    = lane & 15;
  const int n0   = wave * 16;        // each wave owns one 16-wide N tile
  for (int s = 1; s + 1 < path_len; ++s) {
    const int cur = path[s], nxt = path[s + 1];
    const int e = edges[cur];
    const float* W = edge_weights + e * DIM * DIM;
    const float* B = edge_biases + e * DIM;
    v8f cfr = {0.f, 0.f, 0.f, 0.f, 0.f, 0.f, 0.f, 0.f};
    for (int k0 = 0; k0 < DIM; k0 += 4) {
      // A 16x4 f32: lane<16 holds K=k0..k0+1, lane>=16 holds K=k0+2..k0+3
      v2f a, b;
      a.x = ecand[buf][l][k0 + 2 * half];
      a.y = ecand[buf][l][k0 + 2 * half + 1];
      // B 4x16 f32: same K split, N = n0 + l striped across lanes
      b.x = W[(k0 + 2 * half)     * DIM + n0 + l];
      b.y = W[(k0 + 2 * half + 1) * DIM + n0 + l];
      cfr = __builtin_amdgcn_wmma_f32_16x16x4_f32(
          /*neg_a=*/false, a, /*neg_b=*/false, b,
          /*c_mod=*/(short)0, cfr, /*reuse_a=*/false, /*reuse_b=*/false);
    }
    // C VGPR r -> candidate m = r + 8*half, column n = n0 + l
    const int nn = n0 + l;
    const float base = acc[nxt][nn] + B[nn];
#pragma unroll
    for (int r = 0; r < 8; ++r) {
      const int m = r + 8 * half;
      ecand[buf ^ 1][m][nn] = fmaxf(cfr[r] + base, 0.f);  // ancestors all relu
    }
    __syncthreads();
    buf ^= 1;
  }

  // -------- Phase C: scoring --------
  if (tid < EE) {
    float s = 0.f, sd = 0.f;
    for (int n = 0; n < DIM; ++n) {
      s  = fmaf(ecand[buf][tid][n] + seb[n], sew[n], s);
      sd = fmaf(acc[pos][n] + sdb[n], sdw[n], sd);   // acc[pos] == leaf[pos]
    }
    out[j * EE + tid] = s + sd;
  }
}

// ---------------------------------------------------------------------------
extern "C" void kernel_launch(void* const* d_in, const int* in_sizes, int n_in,
                              void* d_out, int out_size, void* d_ws, size_t ws_size,
                              hipStream_t stream) {
  const int*   data         = (const int*)d_in[0];
  const int*   types        = (const int*)d_in[1];
  const int*   graphs       = (const int*)d_in[2];
  const int*   edges        = (const int*)d_in[3];
  const int*   pos          = (const int*)d_in[4];
  const float* data_vecs    = (const float*)d_in[5];
  const float* data_weights = (const float*)d_in[6];
  const float* data_biases  = (const float*)d_in[7];
  const float* edge_weights = (const float*)d_in[8];
  const float* edge_biases  = (const float*)d_in[9];
  const float* sew          = (const float*)d_in[10];
  const float* seb          = (const float*)d_in[11];
  const float* sdw          = (const float*)d_in[12];
  const float* sdb          = (const float*)d_in[13];
  float* out = (float*)d_out;

  const int  N = in_sizes[0];                         // 64
  const int  G = in_sizes[2] / N;                     // 8
  const int  T = in_sizes[7] / DIM;                   // 2
  const long V = (long)in_sizes[5] / ((long)T * VDD); // 100000

  float* leaf_ws = (float*)d_ws;   // N*DIM floats

  leaf_kernel<<<N, DIM, 0, stream>>>(data, types, data_vecs, data_weights,
                                     data_biases, leaf_ws, V);
  tree_kernel<<<G, 256, 0, stream>>>(graphs, edges, pos, leaf_ws,
                                     edge_weights, edge_biases,
                                     sew, seb, sdw, sdb, out);
}